// Attention_11673721111097
// MI455X (gfx1250) — compile-verified
//
#include <hip/hip_runtime.h>

typedef float v2f __attribute__((ext_vector_type(2)));
typedef float v8f __attribute__((ext_vector_type(8)));
typedef unsigned int u32x4 __attribute__((ext_vector_type(4)));
typedef int i32x4 __attribute__((ext_vector_type(4)));
typedef int i32x8 __attribute__((ext_vector_type(8)));

#define DIMC 256
#define NHEAD 8
#define HD 32
#define NTOK 256
#define NGRP 64
#define GH 16
#define GW 16
#define PDIM 16
#define LREL 961  // (2*16-1)^2

#if defined(__has_builtin)
#if __has_builtin(__builtin_amdgcn_tensor_load_to_lds)
#define HAVE_TDM 1
#endif
#endif

__device__ __forceinline__ v8f wmma4(v2f a, v2f b, v8f c) {
  // D = A(16x4) * B(4x16) + C, fp32
  return __builtin_amdgcn_wmma_f32_16x16x4_f32(false, a, false, b, (short)0, c,
                                               false, false);
}

#ifdef HAVE_TDM
// 2D tensor-DMA load: tile (tile_x by tile_y) of 4-byte elements, row stride
// stride0 elements, from global gaddr into contiguous LDS at byte offset
// lds_off. D# layout per CDNA5 ISA 8.3/8.4.
__device__ __forceinline__ void tdm_load_2d(unsigned lds_off, const void* gaddr,
                                            unsigned tile_x, unsigned tile_y,
                                            unsigned long long stride0) {
  unsigned long long ga = (unsigned long long)(uintptr_t)gaddr;
  u32x4 g0;
  g0[0] = 1u;                                   // count=1 (valid), user mode
  g0[1] = lds_off;                              // lds_addr (bytes)
  g0[2] = (unsigned)(ga & 0xFFFFFFFFu);         // global_addr[31:0]
  g0[3] = (unsigned)((ga >> 32) & 0x01FFFFFFu)  // global_addr[56:32]
          | (2u << 30);                         // type=2 ("image")
  i32x8 g1;
  g1[0] = (int)(2u << 16);                      // data_size = 4 bytes
  g1[1] = (int)((tile_x & 0xFFFFu) << 16);      // tensor_dim0[15:0]
  g1[2] = (int)((tile_y & 0xFFFFu) << 16);      // tensor_dim0[31:16]=0 | tensor_dim1[15:0]
  g1[3] = (int)((tile_x & 0xFFFFu) << 16);      // tensor_dim1[31:16]=0 | tile_dim0
  g1[4] = (int)(tile_y & 0xFFFFu);              // tile_dim1 | tile_dim2=0
  g1[5] = (int)(stride0 & 0xFFFFFFFFull);       // tensor_dim0_stride[31:0]
  g1[6] = (int)((stride0 >> 32) & 0xFFFFull);   // stride0[47:32] | stride1=0
  g1[7] = 0;
  i32x4 z4 = {0, 0, 0, 0};
#if __clang_major__ >= 23
  i32x8 z8 = {0, 0, 0, 0, 0, 0, 0, 0};
  __builtin_amdgcn_tensor_load_to_lds(g0, g1, z4, z4, z8, 0);
#else
  __builtin_amdgcn_tensor_load_to_lds(g0, g1, z4, z4, 0);
#endif
}
#endif

// ---------------------------------------------------------------------------
// Kernel 1: DynamicPosBias MLP  -> pos[LREL][NHEAD]
// ---------------------------------------------------------------------------
__device__ __forceinline__ void ln16(float* x, const float* g, const float* b) {
  float m = 0.f;
#pragma unroll
  for (int p = 0; p < PDIM; p++) m += x[p];
  m *= (1.0f / PDIM);
  float v = 0.f;
#pragma unroll
  for (int p = 0; p < PDIM; p++) { float d = x[p] - m; v += d * d; }
  v *= (1.0f / PDIM);
  float r = rsqrtf(v + 1e-5f);
#pragma unroll
  for (int p = 0; p < PDIM; p++) x[p] = (x[p] - m) * r * g[p] + b[p];
}

__global__ void pos_mlp_kernel(const float* __restrict__ pp_w, const float* __restrict__ pp_b,
                               const float* __restrict__ ln1_g, const float* __restrict__ ln1_b,
                               const float* __restrict__ l1_w, const float* __restrict__ l1_b,
                               const float* __restrict__ ln2_g, const float* __restrict__ ln2_b,
                               const float* __restrict__ l2_w, const float* __restrict__ l2_b,
                               const float* __restrict__ ln3_g, const float* __restrict__ ln3_b,
                               const float* __restrict__ l3_w, const float* __restrict__ l3_b,
                               float* __restrict__ pos) {
  int l = blockIdx.x * blockDim.x + threadIdx.x;
  if (l >= LREL) return;
  int i = l / (2 * GW - 1);
  int j = l % (2 * GW - 1);
  float bh = (float)(i - (GH - 1));
  float bw = (float)(j - (GW - 1));
  float x[PDIM], y[PDIM];
#pragma unroll
  for (int p = 0; p < PDIM; p++)
    x[p] = pp_w[2 * p] * bh + pp_w[2 * p + 1] * bw + pp_b[p];
  ln16(x, ln1_g, ln1_b);
#pragma unroll
  for (int p2 = 0; p2 < PDIM; p2++) {
    float s = l1_b[p2];
#pragma unroll
    for (int p = 0; p < PDIM; p++) s += fmaxf(x[p], 0.f) * l1_w[p2 * PDIM + p];
    y[p2] = s;
  }
  ln16(y, ln2_g, ln2_b);
#pragma unroll
  for (int p2 = 0; p2 < PDIM; p2++) {
    float s = l2_b[p2];
#pragma unroll
    for (int p = 0; p < PDIM; p++) s += fmaxf(y[p], 0.f) * l2_w[p2 * PDIM + p];
    x[p2] = s;
  }
  ln16(x, ln3_g, ln3_b);
#pragma unroll
  for (int h = 0; h < NHEAD; h++) {
    float s = l3_b[h];
#pragma unroll
    for (int p = 0; p < PDIM; p++) s += fmaxf(x[p], 0.f) * l3_w[h * PDIM + p];
    pos[l * NHEAD + h] = s;
  }
}

// ---------------------------------------------------------------------------
// Kernel 2: expand pos table -> rpb[NHEAD][NTOK][NTOK]
// ---------------------------------------------------------------------------
__global__ void rpb_kernel(const float* __restrict__ pos, float* __restrict__ rpb) {
  int t = blockIdx.x * 256 + threadIdx.x;  // h*N*N + i*N + j
  int h = t >> 16;
  int ij = t & 65535;
  int i = ij >> 8, j = ij & 255;
  int ih = i >> 4, iw = i & 15, jh = j >> 4, jw = j & 15;
  int idx = (ih - jh + GH - 1) * (2 * GW - 1) + (iw - jw + GW - 1);
  rpb[t] = pos[idx * NHEAD + h];
}

// ---------------------------------------------------------------------------
// Kernel 3: generic fp32 WMMA GEMM:  out[M][Nout] = (A[M][K] @ W[Nout][K]^T + bias)*scale
// One wave -> 16x64 tile; K processed in 16-wide blocks: 20 loads then 16 WMMAs
// so LOADcnt waits amortize over a WMMA burst.
// ---------------------------------------------------------------------------
__global__ void gemm_wmma_kernel(const float* __restrict__ A, const float* __restrict__ W,
                                 const float* __restrict__ bias, float* __restrict__ out,
                                 int M, int K, int Nout, float scale) {
  int lane = threadIdx.x & 31;
  int wid = threadIdx.x >> 5;
  int tile = blockIdx.x * 8 + wid;
  int cgroups = Nout >> 6;
  int rt = tile / cgroups;
  int cg = tile % cgroups;
  if (rt * 16 >= M) return;  // wave-uniform

  int mrow = lane & 15;
  int koff = (lane >> 4) * 2;
  const float* arow = A + (size_t)(rt * 16 + mrow) * K + koff;
  int col0 = cg * 64 + mrow;
  const float* w0 = W + (size_t)(col0) * K + koff;
  const float* w1 = W + (size_t)(col0 + 16) * K + koff;
  const float* w2 = W + (size_t)(col0 + 32) * K + koff;
  const float* w3 = W + (size_t)(col0 + 48) * K + koff;

  v8f acc0 = {0,0,0,0,0,0,0,0}, acc1 = acc0, acc2 = acc0, acc3 = acc0;
  for (int kb = 0; kb < K; kb += 16) {
    __builtin_prefetch(arow + kb + 16, 0, 3);  // next A K-block
    v2f a[4], b0v[4], b1v[4], b2v[4], b3v[4];
#pragma unroll
    for (int s = 0; s < 4; s++) {
      int kk = kb + 4 * s;
      a[s]   = *(const v2f*)(arow + kk);
      b0v[s] = *(const v2f*)(w0 + kk);
      b1v[s] = *(const v2f*)(w1 + kk);
      b2v[s] = *(const v2f*)(w2 + kk);
      b3v[s] = *(const v2f*)(w3 + kk);
    }
#pragma unroll
    for (int s = 0; s < 4; s++) {
      acc0 = wmma4(a[s], b0v[s], acc0);
      acc1 = wmma4(a[s], b1v[s], acc1);
      acc2 = wmma4(a[s], b2v[s], acc2);
      acc3 = wmma4(a[s], b3v[s], acc3);
    }
  }
  int mbase = (lane >> 4) * 8;
  float bb0 = bias[col0], bb1 = bias[col0 + 16], bb2 = bias[col0 + 32], bb3 = bias[col0 + 48];
#pragma unroll
  for (int vr = 0; vr < 8; vr++) {
    size_t row = (size_t)(rt * 16 + mbase + vr);
    float* orow = out + row * Nout + cg * 64 + mrow;
    orow[0]  = (acc0[vr] + bb0) * scale;
    orow[16] = (acc1[vr] + bb1) * scale;
    orow[32] = (acc2[vr] + bb2) * scale;
    orow[48] = (acc3[vr] + bb3) * scale;
  }
}

// ---------------------------------------------------------------------------
// Kernel 4: attention per (batch, head). 8 waves; wave handles 2 q-tiles of 16.
// K/V staged into LDS by the Tensor Data Mover (wave 0), tracked by TENSORcnt.
// LDS: kh 32KB + vh 32KB + per-wave P buffer 8x8KB = 128KB (dynamic).
// ---------------------------------------------------------------------------
__global__ void attn_kernel(const float* __restrict__ qp, const float* __restrict__ kvp,
                            const float* __restrict__ rpb, const float* __restrict__ mask,
                            float* __restrict__ xb) {
  extern __shared__ float smem[];
  float* kh_s = smem;          // [256][32]
  float* vh_s = smem + 8192;   // [256][32]
  float* p_s  = smem + 16384;  // [8 waves][16][128]

  int bh = blockIdx.x;
  int b = bh >> 3;
  int h = bh & 7;
  int g = b & (NGRP - 1);
  int tid = threadIdx.x;
  int lane = tid & 31, wid = tid >> 5;

  const float* ksrc = kvp + ((size_t)b * NTOK) * (2 * DIMC) + h * HD;
#ifdef HAVE_TDM
  if (wid == 0) {
    unsigned kh_off = (unsigned)(uintptr_t)kh_s;  // LDS byte offset (addr[31:0])
    unsigned vh_off = (unsigned)(uintptr_t)vh_s;
    tdm_load_2d(kh_off, ksrc, HD, NTOK, 2 * DIMC);
    tdm_load_2d(vh_off, ksrc + DIMC, HD, NTOK, 2 * DIMC);
    __builtin_amdgcn_s_wait_tensorcnt(0);
  }
#else
  for (int e = tid; e < NTOK * HD; e += 256) {
    int t = e >> 5, d = e & 31;
    kh_s[e] = ksrc[(size_t)t * (2 * DIMC) + d];
    vh_s[e] = ksrc[(size_t)t * (2 * DIMC) + DIMC + d];
  }
#endif
  __syncthreads();

  int mrow = lane & 15;
  int koff = (lane >> 4) * 2;
  int mbase = (lane >> 4) * 8;
  float* pw = p_s + wid * 2048;

  for (int qi = 0; qi < 2; qi++) {
    int qt = wid * 2 + qi;
    // Q A-fragments for all 8 K-steps of hd=32
    v2f af[8];
    const float* qrow = qp + ((size_t)(b * NTOK + qt * 16 + mrow)) * DIMC + h * HD + koff;
#pragma unroll
    for (int s = 0; s < 8; s++) af[s] = *(const v2f*)(qrow + s * 4);

    // S = (Q*scale) K^T + rpb + mask   (scale folded into qp)
    v8f sacc[16];
    for (int ct = 0; ct < 16; ct++) {
      const float* krow = kh_s + (ct * 16 + mrow) * HD + koff;
      v2f bf[8];
#pragma unroll
      for (int s = 0; s < 8; s++) bf[s] = *(const v2f*)(krow + s * 4);
      v8f acc = {0,0,0,0,0,0,0,0};
#pragma unroll
      for (int s = 0; s < 8; s++) acc = wmma4(af[s], bf[s], acc);
      int jcol = ct * 16 + mrow;
#pragma unroll
      for (int vr = 0; vr < 8; vr++) {
        int irow = qt * 16 + mbase + vr;
        acc[vr] += rpb[((size_t)h * NTOK + irow) * NTOK + jcol] +
                   mask[((size_t)g * NTOK + irow) * NTOK + jcol];
      }
      sacc[ct] = acc;
    }

    // row softmax (lanes 0-15 hold rows mbase+0..7, lanes 16-31 rows 8..15)
    float inv[8];
#pragma unroll
    for (int vr = 0; vr < 8; vr++) {
      float m = sacc[0][vr];
#pragma unroll
      for (int ct = 1; ct < 16; ct++) m = fmaxf(m, sacc[ct][vr]);
      for (int off = 1; off < 16; off <<= 1) m = fmaxf(m, __shfl_xor(m, off, 32));
      float s = 0.f;
#pragma unroll
      for (int ct = 0; ct < 16; ct++) {
        float e = __expf(sacc[ct][vr] - m);
        sacc[ct][vr] = e;
        s += e;
      }
      for (int off = 1; off < 16; off <<= 1) s += __shfl_xor(s, off, 32);
      inv[vr] = 1.0f / s;
    }

    // O = P V, staged through per-wave LDS P buffer in two 128-col halves
    v8f o0 = {0,0,0,0,0,0,0,0}, o1 = o0;
    for (int half = 0; half < 2; half++) {
#pragma unroll
      for (int ct = 0; ct < 8; ct++) {
        int cloc = ct * 16 + mrow;
#pragma unroll
        for (int vr = 0; vr < 8; vr++)
          pw[(mbase + vr) * 128 + cloc] = sacc[half * 8 + ct][vr] * inv[vr];
      }
      asm volatile("s_wait_dscnt 0" ::: "memory");  // per-wave LDS RAW
      for (int kk = 0; kk < 128; kk += 16) {
        v2f a4[4], b0v[4], b1v[4];
#pragma unroll
        for (int s = 0; s < 4; s++) {
          int kx = kk + 4 * s;
          a4[s] = *(const v2f*)(pw + mrow * 128 + kx + koff);
          int tok0 = half * 128 + kx + koff;
          b0v[s].x = vh_s[tok0 * HD + mrow];
          b0v[s].y = vh_s[(tok0 + 1) * HD + mrow];
          b1v[s].x = vh_s[tok0 * HD + 16 + mrow];
          b1v[s].y = vh_s[(tok0 + 1) * HD + 16 + mrow];
        }
#pragma unroll
        for (int s = 0; s < 4; s++) {
          o0 = wmma4(a4[s], b0v[s], o0);
          o1 = wmma4(a4[s], b1v[s], o1);
        }
      }
    }
#pragma unroll
    for (int vr = 0; vr < 8; vr++) {
      int tokr = qt * 16 + mbase + vr;
      float* obase = xb + ((size_t)(b * NTOK + tokr)) * DIMC + h * HD;
      obase[mrow] = o0[vr];
      obase[16 + mrow] = o1[vr];
    }
  }
}

// ---------------------------------------------------------------------------
extern "C" void kernel_launch(void* const* d_in, const int* in_sizes, int n_in,
                              void* d_out, int out_size, void* d_ws, size_t ws_size,
                              hipStream_t stream) {
  const float* q      = (const float*)d_in[0];
  const float* k      = (const float*)d_in[1];
  const float* mask   = (const float*)d_in[2];
  const float* q_w    = (const float*)d_in[3];
  const float* q_b    = (const float*)d_in[4];
  const float* kv_w   = (const float*)d_in[5];
  const float* kv_b   = (const float*)d_in[6];
  const float* proj_w = (const float*)d_in[7];
  const float* proj_b = (const float*)d_in[8];
  const float* pp_w   = (const float*)d_in[9];
  const float* pp_b   = (const float*)d_in[10];
  const float* ln1_g  = (const float*)d_in[11];
  const float* ln1_b  = (const float*)d_in[12];
  const float* l1_w   = (const float*)d_in[13];
  const float* l1_b   = (const float*)d_in[14];
  const float* ln2_g  = (const float*)d_in[15];
  const float* ln2_b  = (const float*)d_in[16];
  const float* l2_w   = (const float*)d_in[17];
  const float* l2_b   = (const float*)d_in[18];
  const float* ln3_g  = (const float*)d_in[19];
  const float* ln3_b  = (const float*)d_in[20];
  const float* l3_w   = (const float*)d_in[21];
  const float* l3_b   = (const float*)d_in[22];
  float* out = (float*)d_out;

  float* ws  = (float*)d_ws;
  float* qp  = ws;                  // 65536*256
  float* kvp = qp + 16777216;       // 65536*512
  float* xb  = kvp + 33554432;      // 65536*256
  float* pos = xb + 16777216;       // 961*8 (padded to 8192)
  float* rpb = pos + 8192;          // 8*256*256

  const int M = 256 * 256;          // B*N rows
  const float qscale = 0.17677669529663687f;  // 32^-0.5

  pos_mlp_kernel<<<4, 256, 0, stream>>>(pp_w, pp_b, ln1_g, ln1_b, l1_w, l1_b,
                                        ln2_g, ln2_b, l2_w, l2_b,
                                        ln3_g, ln3_b, l3_w, l3_b, pos);
  rpb_kernel<<<2048, 256, 0, stream>>>(pos, rpb);

  // q projection (scale folded): 4096 row-tiles * 4 col-groups / 8 waves
  gemm_wmma_kernel<<<2048, 256, 0, stream>>>(q, q_w, q_b, qp, M, 256, 256, qscale);
  // kv projection: 4096 * 8 / 8
  gemm_wmma_kernel<<<4096, 256, 0, stream>>>(k, kv_w, kv_b, kvp, M, 256, 512, 1.0f);
  // attention: one block per (b,h), 128KB dynamic LDS
  attn_kernel<<<2048, 256, 131072, stream>>>(qp, kvp, rpb, mask, xb);
  // output projection
  gemm_wmma_kernel<<<2048, 256, 0, stream>>>(xb, proj_w, proj_b, out, M, 256, 256, 1.0f);
}